// VesicaBirth_17136919511710
// MI455X (gfx1250) — compile-verified
//
#include <hip/hip_runtime.h>
#include <hip/hip_bf16.h>
#include <math.h>

// ---------- constants ----------
#define B_     4
#define TU_    4096
#define TM_    2048
#define D_     1024
#define K2D_   2048     // 2*D (K of the GEMM)
#define KTOP_  4        // MAX_BIRTHS
#define M_     (B_ * KTOP_)   // 16 GEMM rows

typedef float v2f __attribute__((ext_vector_type(2)));
typedef float v8f __attribute__((ext_vector_type(8)));

// ---------------------------------------------------------------------------
// Kernel 1: micro_res[b,t] = mean over Tm of r_emer_acc[b,t,:]
// one wave32 per row; 2048 floats/row -> 16 float4 loads/lane, coalesced.
// grid = B*Tu/8 blocks of 256 (8 waves)
// ---------------------------------------------------------------------------
__global__ __launch_bounds__(256) void vb_reduce_mean(
    const float* __restrict__ r, float* __restrict__ micro_res) {
  int wave = (blockIdx.x * 256 + threadIdx.x) >> 5;   // row id, 0..16383
  int lane = threadIdx.x & 31;
  const float4* p = (const float4*)(r + (size_t)wave * TM_);
  float s = 0.0f;
#pragma unroll
  for (int i = 0; i < 16; ++i) {
    float4 v = p[i * 32 + lane];                      // coalesced 512B/wave
    s += v.x + v.y + v.z + v.w;
  }
#pragma unroll
  for (int off = 16; off; off >>= 1) s += __shfl_xor(s, off, 32);
  if (lane == 0) micro_res[wave] = s * (1.0f / (float)TM_);
}

// ---------------------------------------------------------------------------
// Kernel 2: per-batch top-4 of pair_scores with lowest-index tie-break
// grid = B blocks of 256
// ---------------------------------------------------------------------------
__global__ __launch_bounds__(256) void vb_topk4(
    const float* __restrict__ micro_res, const float* __restrict__ threshold_bias,
    int* __restrict__ topk_idx, float* __restrict__ topk_vals) {
  __shared__ float s_score[TU_];      // 4095 used
  __shared__ float s_val[256];
  __shared__ int   s_idx[256];
  int b = blockIdx.x;
  float thr = 0.45f + tanhf(threshold_bias[0]) * 0.2f;
  const float* mr = micro_res + (size_t)b * TU_;
  for (int i = threadIdx.x; i < TU_ - 1; i += 256) {
    float pr = (mr[i] + mr[i + 1]) * 0.5f;
    s_score[i] = (pr > thr) ? pr : 0.0f;              // pair_res * above
  }
  __syncthreads();
  for (int r = 0; r < KTOP_; ++r) {
    float bv = -INFINITY; int bi = 0x7fffffff;
    for (int i = threadIdx.x; i < TU_ - 1; i += 256) {
      float v = s_score[i];
      if (v > bv) { bv = v; bi = i; }                 // first hit wins ties
    }
    s_val[threadIdx.x] = bv; s_idx[threadIdx.x] = bi;
    __syncthreads();
    for (int off = 128; off; off >>= 1) {
      if (threadIdx.x < off) {
        float ov = s_val[threadIdx.x + off]; int oi = s_idx[threadIdx.x + off];
        float cv = s_val[threadIdx.x];       int ci = s_idx[threadIdx.x];
        if (ov > cv || (ov == cv && oi < ci)) { s_val[threadIdx.x] = ov; s_idx[threadIdx.x] = oi; }
      }
      __syncthreads();
    }
    if (threadIdx.x == 0) {
      topk_idx[b * KTOP_ + r]  = s_idx[0];
      topk_vals[b * KTOP_ + r] = s_val[0];
      s_score[s_idx[0]] = -INFINITY;                  // exclude from next round
    }
    __syncthreads();
  }
}

// ---------------------------------------------------------------------------
// Kernel 3: child_raw (16 x 1024) = combined (16 x 2048) @ W^T via
// V_WMMA_F32_16X16X4_F32. One wave per 16-wide N tile, grid = 64 blocks of 32.
// A frag layout (16x4 f32): VGPR0: lanes0-15 K=k0, lanes16-31 K=k0+2; VGPR1: +1.
// B frag layout (4x16 f32) mirrors it with N across lanes.
// ---------------------------------------------------------------------------
__global__ __launch_bounds__(32) void vb_gemm_wmma(
    const float* __restrict__ o_micro, const float* __restrict__ W,
    const int* __restrict__ topk_idx, float* __restrict__ child_raw) {
  int lane = threadIdx.x & 31;
  int n0   = blockIdx.x * 16;
  int m    = lane & 15;                 // GEMM row this lane feeds for A
  int bb   = m >> 2;
  int jj   = m & 3;
  int idx  = topk_idx[bb * KTOP_ + jj];
  const float* pL = o_micro + ((size_t)bb * TU_ + idx) * D_;   // left parent row
  const float* pR = pL + D_;                                   // right parent row
  int n    = n0 + (lane & 15);
  const float* pW = W + (size_t)n * K2D_;                      // W row n (length 2048)
  int koff = (lane < 16) ? 0 : 2;

  v8f c = {};
  for (int k0 = 0; k0 < K2D_; k0 += 4) {
    int f = k0 + koff;                 // even, so f and f+1 are on the same side of 1024
    const float2 av = *(const float2*)((f < D_) ? (pL + f) : (pR + (f - D_)));
    const float2 bv = *(const float2*)(pW + f);
    v2f a;  a.x = av.x;  a.y = av.y;   // VGPR0 = K=f, VGPR1 = K=f+1
    v2f bf; bf.x = bv.x; bf.y = bv.y;
    c = __builtin_amdgcn_wmma_f32_16x16x4_f32(
        /*neg_a=*/false, a, /*neg_b=*/false, bf,
        /*c_mod=*/(short)0, c, /*reuse_a=*/false, /*reuse_b=*/false);
  }
  // C layout: VGPR r -> M=r (lanes 0-15) / M=r+8 (lanes 16-31), N = lane&15
  int mbase = (lane < 16) ? 0 : 8;
#pragma unroll
  for (int r = 0; r < 8; ++r) {
    child_raw[(size_t)(mbase + r) * D_ + n] = c[r];
  }
}

// ---------------------------------------------------------------------------
// Kernel 4: gates[m] = sigmoid(dot(child_raw[m], w_gate) + b_gate)
// one block of 512 = 16 waves, one wave per child row
// ---------------------------------------------------------------------------
__global__ __launch_bounds__(512) void vb_gate(
    const float* __restrict__ child_raw, const float* __restrict__ w_gate,
    const float* __restrict__ b_gate, float* __restrict__ gates) {
  int wave = threadIdx.x >> 5;          // 0..15
  int lane = threadIdx.x & 31;
  const float* row = child_raw + (size_t)wave * D_;
  float s = 0.0f;
#pragma unroll
  for (int i = 0; i < D_ / 32; ++i) {
    int c = i * 32 + lane;
    s += row[c] * w_gate[c];
  }
#pragma unroll
  for (int off = 16; off; off >>= 1) s += __shfl_xor(s, off, 32);
  if (lane == 0) gates[wave] = 1.0f / (1.0f + expf(-(s + b_gate[0])));
}

// ---------------------------------------------------------------------------
// Kernel 5: assemble output.
//  blocks [0, 8192): stream-copy o_macro with row remap (2048 -> 2052 rows/batch)
//  blocks [8192, 8208): gated+masked child rows (topk_vals sorted desc => the
//    stable compaction is the identity permutation, masking suffices)
//  block 8208: n_births scalar
// ---------------------------------------------------------------------------
__global__ __launch_bounds__(256) void vb_assemble(
    const float* __restrict__ o_macro, const float* __restrict__ child_raw,
    const float* __restrict__ gates, const float* __restrict__ topk_vals,
    float* __restrict__ out, int out_size) {
  int bid = blockIdx.x;
  if (bid < 8192) {
    int e   = bid * 256 + threadIdx.x;  // float4 index into o_macro (2M total)
    int row = e >> 8;                   // 256 float4 per 1024-float row
    int col = e & 255;
    int b   = row >> 11;                // 2048 rows per batch
    int t   = row & 2047;
    const float4* src = (const float4*)o_macro;
    float4* dst = (float4*)out;
    dst[((size_t)b * (TM_ + KTOP_) + t) * 256 + col] = src[e];
  } else if (bid < 8192 + M_) {
    int m = bid - 8192;                 // child row 0..15
    int b = m >> 2, j = m & 3;
    float g = gates[m] * ((topk_vals[m] > 0.0f) ? 1.0f : 0.0f);
    const float4* src = (const float4*)(child_raw + (size_t)m * D_);
    float4* dst = (float4*)(out + ((size_t)b * (TM_ + KTOP_) + TM_ + j) * D_);
    float4 v = src[threadIdx.x];
    v.x *= g; v.y *= g; v.z *= g; v.w *= g;
    dst[threadIdx.x] = v;
  } else if (threadIdx.x == 0) {
    int cnt = 0;
#pragma unroll
    for (int i = 0; i < M_; ++i) cnt += (topk_vals[i] > 0.0f) ? 1 : 0;
    out[out_size - 1] = (float)cnt;     // n_births
  }
}

// ---------------------------------------------------------------------------
extern "C" void kernel_launch(void* const* d_in, const int* in_sizes, int n_in,
                              void* d_out, int out_size, void* d_ws, size_t ws_size,
                              hipStream_t stream) {
  const float* o_micro        = (const float*)d_in[0];   // (4,4096,1024) f32
  const float* o_macro        = (const float*)d_in[1];   // (4,2048,1024) f32
  const float* r_emer_acc     = (const float*)d_in[2];   // (4,4096,2048) f32
  /* chunk_size = d_in[3] (unused) */
  const float* W_vesica       = (const float*)d_in[4];   // (1024,2048) f32
  const float* w_gate         = (const float*)d_in[5];   // (1024,) f32
  const float* b_gate         = (const float*)d_in[6];   // scalar
  const float* threshold_bias = (const float*)d_in[7];   // scalar
  float* out = (float*)d_out;

  // workspace layout (bytes)
  char* ws = (char*)d_ws;
  float* micro_res = (float*)(ws);                        // 16384 f = 65536 B
  int*   topk_idx  = (int*)  (ws + 65536);                // 16 i   = 64 B
  float* topk_vals = (float*)(ws + 65600);                // 16 f   = 64 B
  float* child_raw = (float*)(ws + 65664);                // 16*1024 f = 65536 B
  float* gates     = (float*)(ws + 131200);               // 16 f

  // 1) mean over macro dim: 16384 rows, 8 waves/block
  vb_reduce_mean<<<(B_ * TU_) / 8, 256, 0, stream>>>(r_emer_acc, micro_res);

  // 2) per-batch top-4 pair scores
  vb_topk4<<<B_, 256, 0, stream>>>(micro_res, threshold_bias, topk_idx, topk_vals);

  // 3) child projection via f32 WMMA: 64 N-tiles, one wave each
  vb_gemm_wmma<<<D_ / 16, 32, 0, stream>>>(o_micro, W_vesica, topk_idx, child_raw);

  // 4) birth gate
  vb_gate<<<1, 512, 0, stream>>>(child_raw, w_gate, b_gate, gates);

  // 5) output assembly (+ n_births)
  vb_assemble<<<8192 + M_ + 1, 256, 0, stream>>>(o_macro, child_raw, gates,
                                                 topk_vals, out, out_size);
}